// Filter_55113020342611
// MI455X (gfx1250) — compile-verified
//
#include <hip/hip_runtime.h>

typedef __attribute__((ext_vector_type(2))) float v2f;
typedef __attribute__((ext_vector_type(8))) float v8f;

#define N_DE 12288
#define M_X  2048
#define DIM  128
#define BLK  64
#define KC   64
#define LDSS 68   // 64 + 4 pad: rows land on distinct banks, 16B-aligned rows

__device__ __forceinline__ float multi_rbf(float d2) {
    // BANDWIDTHS = (10, 512): -0.5/10 = -0.05 ; -0.5/512 = -0.0009765625
    return __expf(-0.05f * d2) + __expf(-0.0009765625f * d2);
}

// ---------------- row squared-norms: one wave per row ----------------
__global__ __launch_bounds__(256) void row_sq_kernel(const float* __restrict__ A,
                                                     float* __restrict__ sq,
                                                     int nrows) {
    int wave = threadIdx.x >> 5;
    int lane = threadIdx.x & 31;
    int row  = blockIdx.x * 8 + wave;
    if (row >= nrows) return;
    const float4* p4 = reinterpret_cast<const float4*>(A + (size_t)row * DIM);
    float4 v = p4[lane];                       // 32 lanes x 4 floats = 128
    float s = v.x * v.x + v.y * v.y + v.z * v.z + v.w * v.w;
    s += __shfl_xor(s, 16, 32);
    s += __shfl_xor(s, 8, 32);
    s += __shfl_xor(s, 4, 32);
    s += __shfl_xor(s, 2, 32);
    s += __shfl_xor(s, 1, 32);
    if (lane == 0) sq[row] = s;
}

// ---------------- kxx: sum over NxN of multi_rbf(||De_i - De_j||^2) ----------------
__global__ __launch_bounds__(256) void kxx_kernel(const float* __restrict__ De,
                                                  const float* __restrict__ sq_de,
                                                  float* __restrict__ kxx_sum) {
    int bi = blockIdx.x;
    int bj = blockIdx.y;
    if (bi > bj) return;                        // symmetry: keep upper triangle
    __shared__ float As[BLK * LDSS];
    __shared__ float Bs[BLK * LDSS];
    const int i0 = bi * BLK, j0 = bj * BLK;
    const int t = threadIdx.x;
    const int wave = t >> 5;
    const int lane = t & 31;
    const int half = lane >> 4;
    const int l16  = lane & 15;
    const int iloc = (wave & 3) * 16;
    const int jloc = (wave >> 2) * 32;

    v8f c0 = {}; v8f c1 = {};
    const float* arow  = &As[(iloc + l16) * LDSS + 2 * half];
    const float* brow0 = &Bs[(jloc + l16) * LDSS + 2 * half];
    const float* brow1 = &Bs[(jloc + 16 + l16) * LDSS + 2 * half];

    for (int kc = 0; kc < DIM; kc += KC) {
        // cooperative stage: 64 rows x 64 cols into As (rows i0..) and Bs (rows j0..)
        #pragma unroll
        for (int it = 0; it < 4; ++it) {
            int idx = t + it * 256;             // 0..1023 float4 slots
            int row = idx >> 4;                 // 16 float4 per row
            int c4  = (idx & 15) << 2;
            float4 a = *reinterpret_cast<const float4*>(De + (size_t)(i0 + row) * DIM + kc + c4);
            float4 b = *reinterpret_cast<const float4*>(De + (size_t)(j0 + row) * DIM + kc + c4);
            *reinterpret_cast<float4*>(&As[row * LDSS + c4]) = a;
            *reinterpret_cast<float4*>(&Bs[row * LDSS + c4]) = b;
        }
        __syncthreads();
        #pragma unroll
        for (int k = 0; k < KC; k += 4) {
            v2f a  = *reinterpret_cast<const v2f*>(arow + k);
            v2f b0 = *reinterpret_cast<const v2f*>(brow0 + k);
            v2f b1 = *reinterpret_cast<const v2f*>(brow1 + k);
            c0 = __builtin_amdgcn_wmma_f32_16x16x4_f32(false, a, false, b0, (short)0, c0, false, false);
            c1 = __builtin_amdgcn_wmma_f32_16x16x4_f32(false, a, false, b1, (short)0, c1, false, false);
        }
        __syncthreads();
    }

    // epilogue: d2 = sq_i + sq_j - 2*dot, rbf, reduce, one atomic per wave
    int jg = j0 + jloc + l16;
    float sqj0 = sq_de[jg];
    float sqj1 = sq_de[jg + 16];
    float s = 0.0f;
    #pragma unroll
    for (int r = 0; r < 8; ++r) {
        float sqi = sq_de[i0 + iloc + r + 8 * half];
        s += multi_rbf(sqi + sqj0 - 2.0f * c0[r]);
        s += multi_rbf(sqi + sqj1 - 2.0f * c1[r]);
    }
    if (bi != bj) s *= 2.0f;                    // off-diagonal blocks count twice
    s += __shfl_xor(s, 16, 32);
    s += __shfl_xor(s, 8, 32);
    s += __shfl_xor(s, 4, 32);
    s += __shfl_xor(s, 2, 32);
    s += __shfl_xor(s, 1, 32);
    if (lane == 0) atomicAdd(kxx_sum, s);
}

// ---------------- kxy: per-row sums over N of multi_rbf(||X_m - De_j||^2) ----------------
__global__ __launch_bounds__(256) void kxy_kernel(const float* __restrict__ X,
                                                  const float* __restrict__ De,
                                                  const float* __restrict__ sq_x,
                                                  const float* __restrict__ sq_de,
                                                  float* __restrict__ kxy_acc) {
    __shared__ float As[BLK * LDSS];
    __shared__ float Bs[BLK * LDSS];
    const int i0 = blockIdx.x * BLK;            // over M
    const int j0 = blockIdx.y * BLK;            // over N
    const int t = threadIdx.x;
    const int wave = t >> 5;
    const int lane = t & 31;
    const int half = lane >> 4;
    const int l16  = lane & 15;
    const int iloc = (wave & 3) * 16;
    const int jloc = (wave >> 2) * 32;

    v8f c0 = {}; v8f c1 = {};
    const float* arow  = &As[(iloc + l16) * LDSS + 2 * half];
    const float* brow0 = &Bs[(jloc + l16) * LDSS + 2 * half];
    const float* brow1 = &Bs[(jloc + 16 + l16) * LDSS + 2 * half];

    for (int kc = 0; kc < DIM; kc += KC) {
        #pragma unroll
        for (int it = 0; it < 4; ++it) {
            int idx = t + it * 256;
            int row = idx >> 4;
            int c4  = (idx & 15) << 2;
            float4 a = *reinterpret_cast<const float4*>(X  + (size_t)(i0 + row) * DIM + kc + c4);
            float4 b = *reinterpret_cast<const float4*>(De + (size_t)(j0 + row) * DIM + kc + c4);
            *reinterpret_cast<float4*>(&As[row * LDSS + c4]) = a;
            *reinterpret_cast<float4*>(&Bs[row * LDSS + c4]) = b;
        }
        __syncthreads();
        #pragma unroll
        for (int k = 0; k < KC; k += 4) {
            v2f a  = *reinterpret_cast<const v2f*>(arow + k);
            v2f b0 = *reinterpret_cast<const v2f*>(brow0 + k);
            v2f b1 = *reinterpret_cast<const v2f*>(brow1 + k);
            c0 = __builtin_amdgcn_wmma_f32_16x16x4_f32(false, a, false, b0, (short)0, c0, false, false);
            c1 = __builtin_amdgcn_wmma_f32_16x16x4_f32(false, a, false, b1, (short)0, c1, false, false);
        }
        __syncthreads();
    }

    int jg = j0 + jloc + l16;
    float sqj0 = sq_de[jg];
    float sqj1 = sq_de[jg + 16];
    #pragma unroll
    for (int r = 0; r < 8; ++r) {
        int ig = i0 + iloc + r + 8 * half;      // lanes 0-15: row r; lanes 16-31: row r+8
        float sqi = sq_x[ig];
        float v = multi_rbf(sqi + sqj0 - 2.0f * c0[r])
                + multi_rbf(sqi + sqj1 - 2.0f * c1[r]);
        // reduce the 16 N-columns held within each 16-lane half
        v += __shfl_xor(v, 1, 32);
        v += __shfl_xor(v, 2, 32);
        v += __shfl_xor(v, 4, 32);
        v += __shfl_xor(v, 8, 32);
        if (l16 == 0) atomicAdd(&kxy_acc[ig], v);
    }
}

// ---------------- combine: out[m] = kxx_mean + 2 - 2*kxy_mean[m] ----------------
__global__ __launch_bounds__(256) void combine_kernel(const float* __restrict__ ws,
                                                      float* __restrict__ out) {
    int m = blockIdx.x * blockDim.x + threadIdx.x;
    if (m >= M_X) return;
    float kxx_mean = ws[0] * (1.0f / ((float)N_DE * (float)N_DE));
    float kxy_mean = ws[1 + m] * (1.0f / (float)N_DE);
    out[m] = kxx_mean + 2.0f - 2.0f * kxy_mean;
}

__global__ __launch_bounds__(256) void zero_kernel(float* __restrict__ p, int n) {
    int i = blockIdx.x * blockDim.x + threadIdx.x;
    if (i < n) p[i] = 0.0f;
}

extern "C" void kernel_launch(void* const* d_in, const int* in_sizes, int n_in,
                              void* d_out, int out_size, void* d_ws, size_t ws_size,
                              hipStream_t stream) {
    const float* De = (const float*)d_in[0];   // [12288,128] f32
    const float* X  = (const float*)d_in[1];   // [2048,128]  f32
    float* out = (float*)d_out;                // [2048] f32
    float* ws  = (float*)d_ws;

    float* kxx_sum = ws;                       // 1
    float* kxy_acc = ws + 1;                   // M_X
    float* sq_de   = ws + 1 + M_X;             // N_DE
    float* sq_x    = sq_de + N_DE;             // M_X   (total ~64 KB)

    zero_kernel<<<(1 + M_X + 255) / 256, 256, 0, stream>>>(ws, 1 + M_X);
    row_sq_kernel<<<N_DE / 8, 256, 0, stream>>>(De, sq_de, N_DE);
    row_sq_kernel<<<M_X / 8, 256, 0, stream>>>(X, sq_x, M_X);

    dim3 gxx(N_DE / BLK, N_DE / BLK);          // 192 x 192 (lower triangle exits)
    kxx_kernel<<<gxx, 256, 0, stream>>>(De, sq_de, kxx_sum);

    dim3 gxy(M_X / BLK, N_DE / BLK);           // 32 x 192
    kxy_kernel<<<gxy, 256, 0, stream>>>(X, De, sq_x, sq_de, kxy_acc);

    combine_kernel<<<(M_X + 255) / 256, 256, 0, stream>>>(ws, out);
}